// WaveConv2d_40072044872089
// MI455X (gfx1250) — compile-verified
//
#include <hip/hip_runtime.h>

typedef float v2f __attribute__((ext_vector_type(2)));
typedef float v8f __attribute__((ext_vector_type(8)));

#define NPIX   1024     // 32*32
#define NMODE  100      // 50 complex modes as re/im
#define NPAD   112      // padded to multiple of 16
#define BCNT   512      // 8 batch * 64 channels
#define ROWS   2048     // 4 bands * BCNT
#define PLANE  (BCNT * NPIX)   // floats per band plane

// ---------------------------------------------------------------------------
// Build forward DFT matrix F[1024][112] and inverse matrix G[112][1024].
// mode m = 2*(k*5+ky)+c ; k<5 -> kx=k, k>=5 -> kx=k+22 (27..31); c: 0=re,1=im
// Forward (rfft2):  F[p][m] = cos(th) / -sin(th),  th = 2*pi*(kx*x+ky*y)/32
// Inverse (irfft2 of sparse half-spectrum): out = (1/1024)*sum m(ky)*(Re*cos - Im*sin)
// ---------------------------------------------------------------------------
__global__ __launch_bounds__(256) void k_fill_dft(float* __restrict__ F,
                                                  float* __restrict__ G) {
  int idx = blockIdx.x * blockDim.x + threadIdx.x;
  const int total = NPIX * NPAD;
  if (idx >= 2 * total) return;
  bool isG = idx >= total;
  int t = isG ? (idx - total) : idx;
  int m, p;
  if (!isG) { p = t / NPAD; m = t % NPAD; }      // F row-major [p][m]
  else      { m = t / NPIX; p = t % NPIX; }      // G row-major [m][p]
  float val = 0.0f;
  if (m < NMODE) {
    int q = m >> 1, c = m & 1;
    int k = q / 5, ky = q % 5;
    int kx = (k < 5) ? k : (k + 22);
    int x = p >> 5, y = p & 31;
    int ph = (kx * x + ky * y) & 31;
    float th = (float)ph * 0.19634954084936207f;   // 2*pi/32
    float cs = cosf(th), sn = sinf(th);
    if (!isG) {
      val = (c == 0) ? cs : -sn;
    } else {
      float s = ((ky == 0) ? 1.0f : 2.0f) * (1.0f / 1024.0f);
      val = (c == 0) ? s * cs : -s * sn;
    }
  }
  if (!isG) F[t] = val; else G[t] = val;
}

// ---------------------------------------------------------------------------
// Fused 3-level Haar DWT: each thread reduces one 8x8 block of x into the
// four level-3 coefficients (LL, LH, HL, HH), each scaled by 1/8.
// ---------------------------------------------------------------------------
__global__ __launch_bounds__(256) void k_dwt_fwd(const float* __restrict__ x,
                                                 float* __restrict__ coeff) {
  int bc = blockIdx.x;                         // 0..511  (b*64+c)
  int j  = threadIdx.x & 31;                   // output col 0..31
  int i  = blockIdx.y * 8 + (threadIdx.x >> 5);// output row 0..31
  const float* base = x + (size_t)bc * 65536 + (size_t)(i * 8) * 256 + j * 8;
  float q00 = 0.f, q01 = 0.f, q10 = 0.f, q11 = 0.f;
#pragma unroll
  for (int r = 0; r < 8; ++r) {
    float4 lo = *(const float4*)(base + r * 256);
    float4 hi = *(const float4*)(base + r * 256 + 4);
    float sl = (lo.x + lo.y) + (lo.z + lo.w);
    float sr = (hi.x + hi.y) + (hi.z + hi.w);
    if (r < 4) { q00 += sl; q01 += sr; } else { q10 += sl; q11 += sr; }
  }
  float a = q00 + q01, b = q10 + q11, c = q00 - q01, d = q10 - q11;
  int o = bc * NPIX + i * 32 + j;
  coeff[0 * PLANE + o] = (a + b) * 0.125f;   // LL
  coeff[1 * PLANE + o] = (a - b) * 0.125f;   // LH
  coeff[2 * PLANE + o] = (c + d) * 0.125f;   // HL
  coeff[3 * PLANE + o] = (c - d) * 0.125f;   // HH
}

// ---------------------------------------------------------------------------
// f32 WMMA GEMM: C[M][N] = A[M][K] @ B[K][N], one 16x16 tile per wave32,
// K-loop of V_WMMA_F32_16X16X4_F32. M,N multiples of 16; K multiple of 4.
// A frag (16x4): lane L holds A[L&15][k0 + 2*(L>>4) + {0,1}]
// B frag (4x16): lane L holds B[k0 + 2*(L>>4) + {0,1}][L&15]
// C/D: vgpr r, lane L -> row r + 8*(L>>4), col L&15
// ---------------------------------------------------------------------------
__global__ __launch_bounds__(128) void k_gemm_wmma(const float* __restrict__ A,
                                                   const float* __restrict__ B,
                                                   float* __restrict__ C,
                                                   int M, int N, int K) {
  int wave = blockIdx.x * (blockDim.x >> 5) + (threadIdx.x >> 5);
  int lane = threadIdx.x & 31;
  int ntl = N >> 4;
  int mt = wave / ntl;
  int nt = wave - mt * ntl;
  if (mt * 16 >= M) return;
  int row = lane & 15;
  int kh  = (lane >> 4) << 1;                 // 0 or 2
  const float* ap = A + (size_t)(mt * 16 + row) * K;
  const float* bp = B + nt * 16 + row;        // "row" doubles as B column
  v8f acc = {};
  for (int k0 = 0; k0 < K; k0 += 4) {
    v2f av, bv;
    av.x = ap[k0 + kh];
    av.y = ap[k0 + kh + 1];
    bv.x = bp[(size_t)(k0 + kh) * N];
    bv.y = bp[(size_t)(k0 + kh + 1) * N];
    acc = __builtin_amdgcn_wmma_f32_16x16x4_f32(false, av, false, bv,
                                                (short)0, acc, false, false);
  }
  float* cp = C + (size_t)(mt * 16 + 8 * (lane >> 4)) * N + nt * 16 + (lane & 15);
#pragma unroll
  for (int r = 0; r < 8; ++r) cp[(size_t)r * N] = acc[r];
}

// ---------------------------------------------------------------------------
// Complex channel mix per (band, mode): Y[b][o] = sum_i X[b][i] * W[i][o]
// W = w1 (kx 0..4) or w2 (kx 27..31) for the band; complex64 interleaved.
// One block per (band, mode); 512 threads = (b,o); X staged in LDS.
// ---------------------------------------------------------------------------
__global__ __launch_bounds__(512) void k_mix(const float* __restrict__ modes,
                                             float* __restrict__ mixed,
                                             const float* wa1, const float* wa2,
                                             const float* wh1, const float* wh2,
                                             const float* wv1, const float* wv2,
                                             const float* wd1, const float* wd2) {
  __shared__ float xs[8 * 64 * 2];
  int band = blockIdx.x / 50;
  int q    = blockIdx.x % 50;                  // complex mode index
  int k = q / 5, ky = q % 5;
  int t = threadIdx.x;
  for (int e = t; e < 8 * 64 * 2; e += 512) {
    int b = e >> 7; int rem = e & 127; int i = rem >> 1; int c = rem & 1;
    xs[e] = modes[(size_t)(band * 512 + b * 64 + i) * NPAD + 2 * q + c];
  }
  __syncthreads();
  const float* w;
  if      (band == 0) w = (k < 5) ? wa1 : wa2;
  else if (band == 1) w = (k < 5) ? wh1 : wh2;
  else if (band == 2) w = (k < 5) ? wv1 : wv2;
  else                w = (k < 5) ? wd1 : wd2;
  int mx = (k < 5) ? k : (k - 5);
  int o = t & 63, b = t >> 6;
  float ar = 0.f, ai = 0.f;
  for (int i = 0; i < 64; ++i) {
    float xr = xs[b * 128 + 2 * i], xi = xs[b * 128 + 2 * i + 1];
    const float* wp = w + 2 * ((size_t)((i * 64 + o) * 5 + mx) * 5 + ky);
    float wr = wp[0], wi = wp[1];
    ar += xr * wr - xi * wi;
    ai += xr * wi + xi * wr;
  }
  size_t rowb = (size_t)(band * 512 + b * 64 + o) * NPAD;
  mixed[rowb + 2 * q]     = ar;
  mixed[rowb + 2 * q + 1] = ai;
  if (q == 0) {                                // zero pad columns once per row
    for (int m = NMODE; m < NPAD; ++m) mixed[rowb + m] = 0.f;
  }
}

// ---------------------------------------------------------------------------
// Inverse DWT with zero fine details: out[8i+4ri+u][8j+4rj+v] =
//   (LL + (1-2ri)LH + (1-2rj)HL + (1-2ri)(1-2rj)HH) / 8, broadcast over 4x4.
// Each thread writes one float4 (exactly one rj group).
// ---------------------------------------------------------------------------
__global__ __launch_bounds__(256) void k_idwt(const float* __restrict__ outc,
                                              float* __restrict__ out) {
  int bc = blockIdx.x;
  int t4 = threadIdx.x & 63;                   // float4 index in the row
  int X  = blockIdx.y * 4 + (threadIdx.x >> 6);
  int i = X >> 3, ri = (X >> 2) & 1;
  int j = t4 >> 1, rj = t4 & 1;
  int o = bc * NPIX + i * 32 + j;
  float ll = outc[0 * PLANE + o];
  float lh = outc[1 * PLANE + o];
  float hl = outc[2 * PLANE + o];
  float hh = outc[3 * PLANE + o];
  float sh = ri ? -1.f : 1.f;
  float sv = rj ? -1.f : 1.f;
  float v = (ll + sh * lh + sv * hl + sh * sv * hh) * 0.125f;
  float4 vv = {v, v, v, v};
  *(float4*)(out + (size_t)bc * 65536 + (size_t)X * 256 + 4 * t4) = vv;
}

extern "C" void kernel_launch(void* const* d_in, const int* in_sizes, int n_in,
                              void* d_out, int out_size, void* d_ws, size_t ws_size,
                              hipStream_t stream) {
  const float* x = (const float*)d_in[0];
  float* out = (float*)d_out;
  float* ws = (float*)d_ws;

  float* coeff = ws;                    // 4 * 512 * 1024      = 2,097,152 f
  float* outc  = coeff + 4 * PLANE;     // 2,097,152 f
  float* F     = outc + 4 * PLANE;      // 1024*112 = 114,688 f
  float* G     = F + NPIX * NPAD;       // 112*1024 = 114,688 f
  float* modes = G + NPAD * NPIX;       // 2048*112 = 229,376 f
  float* mixed = modes + ROWS * NPAD;   // 229,376 f   (total ~19.5 MB)

  // 1) DFT matrices
  k_fill_dft<<<(2 * NPIX * NPAD + 255) / 256, 256, 0, stream>>>(F, G);
  // 2) forward DWT (reads x once, 134 MB)
  k_dwt_fwd<<<dim3(512, 4), 256, 0, stream>>>(x, coeff);
  // 3) forward truncated DFT: [2048x1024] @ [1024x112], 896 tiles
  k_gemm_wmma<<<224, 128, 0, stream>>>(coeff, F, modes, ROWS, NPAD, NPIX);
  // 4) complex channel mix per (band, mode)
  k_mix<<<4 * 50, 512, 0, stream>>>(modes, mixed,
      (const float*)d_in[1], (const float*)d_in[2],
      (const float*)d_in[3], (const float*)d_in[4],
      (const float*)d_in[5], (const float*)d_in[6],
      (const float*)d_in[7], (const float*)d_in[8]);
  // 5) inverse sparse-mode DFT: [2048x112] @ [112x1024], 8192 tiles
  k_gemm_wmma<<<2048, 128, 0, stream>>>(mixed, G, outc, ROWS, NPIX, NPAD);
  // 6) inverse DWT (writes out once, 134 MB)
  k_idwt<<<dim3(512, 64), 256, 0, stream>>>(outc, out);
}